// NeighborSearch_30605936951526
// MI455X (gfx1250) — compile-verified
//
#include <hip/hip_runtime.h>

typedef float v2f __attribute__((ext_vector_type(2)));
typedef float v4f __attribute__((ext_vector_type(4)));
typedef float v8f __attribute__((ext_vector_type(8)));

#define N_DATA   16384
#define N_QUERY  16384
#define CHUNK    2048          // data points staged in LDS per pass (32 KB as float4)
#define WAVES_PB 8             // waves per block (wave32 -> 256 threads)
#define BLOCK_T  256

// -------- helpers ---------------------------------------------------------

// Build the A operand (16x4 f32, one wave's 16 queries padded with ||q||^2).
// ISA layout: lanes 0-15 -> {VGPR0:K0, VGPR1:K1}; lanes 16-31 -> {VGPR0:K2, VGPR1:K3}
__device__ inline v2f make_A(const float* __restrict__ queries, int mtile,
                             int l16, int half) {
    const int m = mtile * 16 + l16;
    const float qx = queries[3 * m + 0];
    const float qy = queries[3 * m + 1];
    const float qz = queries[3 * m + 2];
    const float q2 = qx * qx + qy * qy + qz * qz;
    v2f a;
    a.x = half ? qz : qx;   // K2 : K0
    a.y = half ? q2 : qy;   // K3 : K1
    return a;
}

// Compute one 16x16 tile of squared distances with a single f32 WMMA.
// v = (x, y, z, ||d||^2) for this lane's column.
__device__ inline v8f tile_sqdist(v2f a, v4f v, int half) {
    v2f b;
    b.x = half ? (-2.0f * v.z) : (-2.0f * v.x);  // K2 : K0
    b.y = half ? 1.0f          : (-2.0f * v.y);  // K3 : K1
    v8f c;
#pragma unroll
    for (int r = 0; r < 8; ++r) c[r] = v.w;      // + ||d||^2 broadcast down the column
    // D = A(16x4) x B(4x16) + C  ->  ||q_m - d_n||^2
    c = __builtin_amdgcn_wmma_f32_16x16x4_f32(false, a, false, b,
                                              (short)0, c, false, false);
    return c;
}

// -------- phase 1: per-query neighbor counts ------------------------------

__global__ __launch_bounds__(BLOCK_T) void nbr_count_kernel(
    const float* __restrict__ data, const float* __restrict__ queries,
    const float* __restrict__ radius, int* __restrict__ counts) {
    __shared__ v4f lds[CHUNK];

    const int lane = threadIdx.x & 31;
    const int wave = threadIdx.x >> 5;
    const int half = lane >> 4;
    const int l16  = lane & 15;
    const float r  = radius[0];
    const float r2 = r * r;

    const int mtile = blockIdx.x * WAVES_PB + wave;
    const v2f a = make_A(queries, mtile, l16, half);

    int cnt[8];
#pragma unroll
    for (int q = 0; q < 8; ++q) cnt[q] = 0;

    for (int base = 0; base < N_DATA; base += CHUNK) {
        __syncthreads();
#pragma unroll
        for (int k = 0; k < CHUNK / BLOCK_T; ++k) {
            const int p = threadIdx.x + k * BLOCK_T;
            const float* d = &data[3 * (base + p)];
            const float x = d[0], y = d[1], z = d[2];
            v4f v; v.x = x; v.y = y; v.z = z; v.w = x * x + y * y + z * z;
            lds[p] = v;
        }
        __syncthreads();

        for (int t = 0; t < CHUNK / 16; ++t) {
            const v4f v = lds[t * 16 + l16];
            const v8f c = tile_sqdist(a, v, half);
#pragma unroll
            for (int q = 0; q < 8; ++q) cnt[q] += (c[q] <= r2) ? 1 : 0;
        }
    }

    // Reduce each row counter across the 16 lanes of its half (xor 1,2,4,8
    // never crosses the 16-lane boundary).
#pragma unroll
    for (int q = 0; q < 8; ++q) {
        int v = cnt[q];
#pragma unroll
        for (int ofs = 1; ofs <= 8; ofs <<= 1) v += __shfl_xor(v, ofs, 32);
        cnt[q] = v;
    }
    if (l16 == 0) {
#pragma unroll
        for (int q = 0; q < 8; ++q) counts[mtile * 16 + half * 8 + q] = cnt[q];
    }
}

// -------- phase 2: scan counts -> offsets (ws) + splits (tail of d_out) ---

__global__ __launch_bounds__(BLOCK_T) void nbr_scan_kernel(
    const int* __restrict__ counts, int* __restrict__ offsets,
    long long* __restrict__ out) {
    __shared__ int partial[BLOCK_T];
    const int t   = threadIdx.x;
    const int PER = N_QUERY / BLOCK_T;  // 64

    int sum = 0;
    for (int i = 0; i < PER; ++i) sum += counts[t * PER + i];
    partial[t] = sum;
    __syncthreads();

    for (int ofs = 1; ofs < BLOCK_T; ofs <<= 1) {
        const int v = partial[t];
        const int w = (t >= ofs) ? partial[t - ofs] : 0;
        __syncthreads();
        partial[t] = v + w;
        __syncthreads();
    }

    const int total = partial[BLOCK_T - 1];
    int run = partial[t] - sum;  // exclusive prefix at start of this thread's span
    for (int i = 0; i < PER; ++i) {
        const int idx = t * PER + i;
        offsets[idx] = run;
        out[(long long)total + idx] = (long long)run;  // splits[idx]
        run += counts[idx];
    }
    if (t == 0) {
        offsets[N_QUERY] = total;
        out[(long long)total + N_QUERY] = (long long)total;  // splits[M]
    }
}

// -------- phase 3: recompute mask, compact neighbor indices ---------------

__global__ __launch_bounds__(BLOCK_T) void nbr_write_kernel(
    const float* __restrict__ data, const float* __restrict__ queries,
    const float* __restrict__ radius, const int* __restrict__ offsets,
    long long* __restrict__ out) {
    __shared__ v4f lds[CHUNK];

    const int lane = threadIdx.x & 31;
    const int wave = threadIdx.x >> 5;
    const int half = lane >> 4;
    const int l16  = lane & 15;
    const float r  = radius[0];
    const float r2 = r * r;

    const int mtile = blockIdx.x * WAVES_PB + wave;
    const v2f a = make_A(queries, mtile, l16, half);

    int cur[8];  // running write cursor per owned row (uniform across each half)
#pragma unroll
    for (int q = 0; q < 8; ++q) cur[q] = offsets[mtile * 16 + half * 8 + q];

    for (int base = 0; base < N_DATA; base += CHUNK) {
        __syncthreads();
#pragma unroll
        for (int k = 0; k < CHUNK / BLOCK_T; ++k) {
            const int p = threadIdx.x + k * BLOCK_T;
            const float* d = &data[3 * (base + p)];
            const float x = d[0], y = d[1], z = d[2];
            v4f v; v.x = x; v.y = y; v.z = z; v.w = x * x + y * y + z * z;
            lds[p] = v;
        }
        __syncthreads();

        for (int t = 0; t < CHUNK / 16; ++t) {
            const v4f v = lds[t * 16 + l16];
            const v8f c = tile_sqdist(a, v, half);
            const int n = base + t * 16 + l16;  // this lane's data index
#pragma unroll
            for (int q = 0; q < 8; ++q) {
                const bool pred = (c[q] <= r2);
                const unsigned bal = __builtin_amdgcn_ballot_w32(pred);
                const unsigned hm  = (bal >> (half * 16)) & 0xFFFFu;
                const int pos = __builtin_popcount(hm & ((1u << l16) - 1u));
                if (pred) out[cur[q] + pos] = (long long)n;
                cur[q] += __builtin_popcount(hm);
            }
        }
    }
}

// -------- launcher --------------------------------------------------------

extern "C" void kernel_launch(void* const* d_in, const int* in_sizes, int n_in,
                              void* d_out, int out_size, void* d_ws, size_t ws_size,
                              hipStream_t stream) {
    (void)in_sizes; (void)n_in; (void)out_size; (void)ws_size;
    const float* data    = (const float*)d_in[0];
    const float* queries = (const float*)d_in[1];
    const float* radius  = (const float*)d_in[2];
    long long*   out     = (long long*)d_out;

    int* counts  = (int*)d_ws;            // N_QUERY ints
    int* offsets = counts + N_QUERY;      // N_QUERY+1 ints

    const dim3 grid(N_QUERY / (16 * WAVES_PB));  // 128 blocks of 8 waves
    nbr_count_kernel<<<grid, BLOCK_T, 0, stream>>>(data, queries, radius, counts);
    nbr_scan_kernel<<<1, BLOCK_T, 0, stream>>>(counts, offsets, out);
    nbr_write_kernel<<<grid, BLOCK_T, 0, stream>>>(data, queries, radius, offsets, out);
}